// MambaBlock_70609262346497
// MI455X (gfx1250) — compile-verified
//
#include <hip/hip_runtime.h>
#include <hip/hip_bf16.h>
#include <math.h>

// Problem constants (from reference)
#define B_SZ     2
#define SEQ      2048
#define D_MODEL  1024
#define D_STATE  16
#define D_DISCR  64
#define KSIZE    4
#define M_ROWS   (B_SZ * SEQ)          // 4096
#define LN_EPS   1e-5f

typedef float v2f __attribute__((ext_vector_type(2)));
typedef float v8f __attribute__((ext_vector_type(8)));
typedef unsigned int v4u __attribute__((ext_vector_type(4)));
typedef int v4i __attribute__((ext_vector_type(4)));
typedef int v8i __attribute__((ext_vector_type(8)));

// Use the Tensor Data Mover when the toolchain exposes the builtin.
#if defined(__HIP_DEVICE_COMPILE__) && defined(__gfx1250__) && \
    __has_builtin(__builtin_amdgcn_tensor_load_to_lds) && \
    __has_builtin(__builtin_amdgcn_s_wait_tensorcnt)
#define USE_TDM 1
#else
#define USE_TDM 0
#endif

__device__ __forceinline__ float silu_f(float v) {
    return v / (1.0f + __expf(-v));
}
__device__ __forceinline__ float softplus_f(float v) {
    return (v > 20.0f) ? v : log1pf(__expf(v));
}

#if USE_TDM
// ---------------------------------------------------------------------------
// Issue one TDM 2-D tile load: tile_y rows of tile_x fp32 elements from a
// row-major tensor with row stride `stride_elems`, deposited into LDS at
// byte offset lds_off with `pad_amount` dwords of padding after each
// 2^(pad_interval_code+1) dwords (used to build the bank-conflict-free
// stride-20 / stride-80 LDS layouts directly from the DMA).
// D# bitfields per CDNA5 ISA.
// This toolchain exposes the 6-arg builtin:
//   (uint32x4 g0, int32x8 g1, int32x4 g2, int32x4 g3, int32x8 extra, i32 cpol)
// ---------------------------------------------------------------------------
__device__ __forceinline__ void tdm_load_2d(const float* gptr, unsigned lds_off,
                                            int tile_x, int tile_y, int stride_elems,
                                            int pad_interval_code, int pad_amount_code)
{
    const unsigned long long ga = (unsigned long long)(uintptr_t)gptr;
    v4u g0;
    g0[0] = 1u;                                           // count=1 (valid user D#)
    g0[1] = lds_off;                                      // lds_addr (bytes)
    g0[2] = (unsigned)ga;                                 // global_addr[31:0]
    g0[3] = (unsigned)((ga >> 32) & 0x01ffffffu)          // global_addr[56:32]
          | (2u << 30);                                   // type=2 ("image")
    v8i g1;
    g1[0] = (2 << 16)                                     // data_size = 4 bytes
          | (1 << 20)                                     // pad_enable
          | (pad_interval_code << 22)
          | (pad_amount_code << 25);
    g1[1] = (tile_x & 0xffff) << 16;                      // tensor_dim0[15:0] (= tile_x)
    g1[2] = (tile_y & 0xffff) << 16;                      // dim0 hi=0 | tensor_dim1[15:0]
    g1[3] = (tile_x << 16);                               // dim1 hi=0 | tile_dim0
    g1[4] = tile_y;                                       // tile_dim1 | tile_dim2=0
    g1[5] = stride_elems;                                 // tensor_dim0_stride[31:0]
    g1[6] = 0;                                            // stride0 hi | stride1 lo
    g1[7] = 0;
    const v4i z4 = {0, 0, 0, 0};                          // groups 2/3: 2-D tile
    const v8i z8 = {0, 0, 0, 0, 0, 0, 0, 0};
    __builtin_amdgcn_tensor_load_to_lds(g0, g1, z4, z4, z8, 0);
}
#endif

// ---------------------------------------------------------------------------
// LDS-tiled fp32 GEMM using V_WMMA_F32_16X16X4_F32.
// Workgroup = 256 threads (8 wave32s) computes a 64x64 tile of C.
// Waves laid out 4(M) x 2(N); each wave owns a 16x32 strip (two 16x16 tiles).
// K staged through double-buffered LDS in 16-wide chunks, filled by the TDM
// (wave 0 programs the descriptor; TENSORcnt + barrier publishes the tile)
// or, on toolchains without the TDM builtin, by per-thread float4 staging.
// ---------------------------------------------------------------------------
template <int EPI>
__global__ __launch_bounds__(256)
void wmma_gemm_tiled_kernel(const float* __restrict__ A,
                            const float* __restrict__ B,
                            float* __restrict__ C,
                            const float* __restrict__ dvec,
                            int K, int lda, int ldb, int ldc)
{
    // Padded strides -> conflict-free fragment reads (64 banks):
    // As stride 20: banks {20m+k} distinct for m=0..15; halves differ by mod-4 class
    // Bs stride 80: the two lane-halves (rows k, k+2) land 32 banks apart
    __shared__ float As[2][64][20];
    __shared__ float Bs[2][16][80];

    const int tid  = threadIdx.x;
    const int lane = tid & 31;
    const int wv   = tid >> 5;          // 0..7
    const int wr   = wv & 3;            // M sub-tile (16 rows each)
    const int wc   = wv >> 2;           // N 32-col group (0/1)
    const int half = lane >> 4;         // WMMA lane half
    const int l16  = lane & 15;

    const int m0 = blockIdx.y * 64;
    const int n0 = blockIdx.x * 64;
    const int nkb = K >> 4;

#if USE_TDM
    const float* Ag0 = A + (size_t)m0 * lda;              // + kb*16 per chunk
    const float* Bg0 = B + n0;                            // + kb*16*ldb per chunk
    if (wv == 0) {
        tdm_load_2d(Ag0, (unsigned)(uintptr_t)&As[0][0][0], 16, 64, lda, 3, 3);
        tdm_load_2d(Bg0, (unsigned)(uintptr_t)&Bs[0][0][0], 64, 16, ldb, 5, 15);
        __builtin_amdgcn_s_wait_tensorcnt(0);
    }
    __syncthreads();
#else
    const int ar = tid >> 2;            // 0..63  (A row)
    const int ac = (tid & 3) * 4;       // 0,4,8,12
    const int br = tid >> 4;            // 0..15  (B row)
    const int bc = (tid & 15) * 4;      // 0..60
    const float* Agp = A + (size_t)(m0 + ar) * lda + ac;
    const float* Bgp = B + (size_t)br * ldb + n0 + bc;
    float4 aq = *(const float4*)Agp;
    float4 bq = *(const float4*)Bgp;
    *(float4*)&As[0][ar][ac] = aq;
    *(float4*)&Bs[0][br][bc] = bq;
    __syncthreads();
#endif

    v8f acc0 = {};
    v8f acc1 = {};

    for (int kb = 0; kb < nkb; ++kb) {
        const int cur = kb & 1;
        const int nxt = cur ^ 1;
#if USE_TDM
        if (kb + 1 < nkb && wv == 0) {   // DMA next chunk while everyone computes
            tdm_load_2d(Ag0 + (size_t)(kb + 1) * 16,
                        (unsigned)(uintptr_t)&As[nxt][0][0], 16, 64, lda, 3, 3);
            tdm_load_2d(Bg0 + (size_t)(kb + 1) * 16 * ldb,
                        (unsigned)(uintptr_t)&Bs[nxt][0][0], 64, 16, ldb, 5, 15);
        }
#else
        if (kb + 1 < nkb) {
            aq = *(const float4*)(Agp + (size_t)(kb + 1) * 16);
            bq = *(const float4*)(Bgp + (size_t)(kb + 1) * 16 * ldb);
        }
#endif
#pragma unroll
        for (int k = 0; k < 16; k += 4) {
            v2f av = *(const v2f*)&As[cur][wr * 16 + l16][k + half * 2];
            v2f bv0, bv1;
            bv0.x = Bs[cur][k + half * 2    ][wc * 32 + l16];
            bv0.y = Bs[cur][k + half * 2 + 1][wc * 32 + l16];
            bv1.x = Bs[cur][k + half * 2    ][wc * 32 + 16 + l16];
            bv1.y = Bs[cur][k + half * 2 + 1][wc * 32 + 16 + l16];
            acc0 = __builtin_amdgcn_wmma_f32_16x16x4_f32(
                false, av, false, bv0, (short)0, acc0, false, false);
            acc1 = __builtin_amdgcn_wmma_f32_16x16x4_f32(
                false, av, false, bv1, (short)0, acc1, false, false);
        }
        if (kb + 1 < nkb) {
#if USE_TDM
            if (wv == 0) __builtin_amdgcn_s_wait_tensorcnt(0);
            __syncthreads();             // publish nxt to all waves
#else
            __syncthreads();             // everyone done reading before refill
            *(float4*)&As[nxt][ar][ac] = aq;
            *(float4*)&Bs[nxt][br][bc] = bq;
            __syncthreads();
#endif
        }
    }

    // C lane layout: VGPR r -> row = half*8 + r, col = l16
#pragma unroll
    for (int r = 0; r < 8; ++r) {
        const int orow = m0 + wr * 16 + half * 8 + r;
        const int col0 = n0 + wc * 32 + l16;
        const int col1 = col0 + 16;
        float v0 = acc0[r];
        float v1 = acc1[r];
        if (EPI == 1) {
            v0 = softplus_f(dvec[col0] + v0);
            v1 = softplus_f(dvec[col1] + v1);
        }
        C[(size_t)orow * ldc + col0] = v0;
        C[(size_t)orow * ldc + col1] = v1;
    }
}

// ---------------------------------------------------------------------------
// Simple one-wave GEMM (for the tiny N=16 projections x@W_sB / x@W_sC).
// ---------------------------------------------------------------------------
__global__ __launch_bounds__(32)
void wmma_gemm_small_kernel(const float* __restrict__ A,
                            const float* __restrict__ B,
                            float* __restrict__ C,
                            int K, int lda, int ldb, int ldc)
{
    const int lane = threadIdx.x & 31;
    const int half = lane >> 4;
    const int l16  = lane & 15;
    const int ti   = blockIdx.y;
    const int tj   = blockIdx.x;

    const int row = ti * 16 + l16;
    const int col = tj * 16 + l16;

    const float* Ap = A + (size_t)row * lda + half * 2;
    const float* Bp = B + (size_t)(half * 2) * ldb + col;

    v8f acc = {};
#pragma unroll 4
    for (int kk = 0; kk < K; kk += 4) {
        v2f av, bv;
        av.x = Ap[kk];
        av.y = Ap[kk + 1];
        bv.x = Bp[(size_t)kk * ldb];
        bv.y = Bp[(size_t)(kk + 1) * ldb];
        acc = __builtin_amdgcn_wmma_f32_16x16x4_f32(
            false, av, false, bv, (short)0, acc, false, false);
    }
#pragma unroll
    for (int r = 0; r < 8; ++r) {
        const int orow = ti * 16 + half * 8 + r;
        C[(size_t)orow * ldc + col] = acc[r];
    }
}

// ---------------------------------------------------------------------------
// Depthwise causal conv (K=4) + bias + SiLU.  a = first D_MODEL cols of ab.
// ---------------------------------------------------------------------------
__global__ __launch_bounds__(256)
void conv_silu_kernel(const float* __restrict__ ab,
                      const float* __restrict__ w,
                      const float* __restrict__ bias,
                      float* __restrict__ x)
{
    const int idx = blockIdx.x * blockDim.x + threadIdx.x;   // over B*L*D
    const int d = idx % D_MODEL;
    const int t = (idx / D_MODEL) % SEQ;
    const int b = idx / (D_MODEL * SEQ);

    const float* a = ab + (size_t)b * SEQ * (2 * D_MODEL) + d; // stride 2D rows
    float acc = bias[d];
#pragma unroll
    for (int j = 0; j < KSIZE; ++j) {
        const int tt = t - (KSIZE - 1) + j;
        if (tt >= 0) acc = fmaf(a[(size_t)tt * (2 * D_MODEL)], w[d * KSIZE + j], acc);
    }
    x[idx] = silu_f(acc);
}

// ---------------------------------------------------------------------------
// Fused selective-scan, software pipelined.
// Block = 256 threads = 16 d-channels x 16 states; grid = (D/16, B).
// ---------------------------------------------------------------------------
__global__ __launch_bounds__(256)
void ssm_scan_kernel(const float* __restrict__ x,
                     const float* __restrict__ s,
                     const float* __restrict__ Bm,
                     const float* __restrict__ Cm,
                     const float* __restrict__ A,
                     float* __restrict__ y)
{
    const int n  = threadIdx.x & 15;          // state index
    const int dl = threadIdx.x >> 4;          // local channel 0..15
    const int d  = blockIdx.x * 16 + dl;
    const int b  = blockIdx.y;

    const float expA = __expf(-A[d * D_STATE + n]);

    const float* xp = x  + (size_t)b * SEQ * D_MODEL + d;
    const float* sp = s  + (size_t)b * SEQ * D_MODEL + d;
    const float* Bp = Bm + (size_t)b * SEQ * D_STATE + n;
    const float* Cp = Cm + (size_t)b * SEQ * D_STATE + n;
    float*       yp = y  + (size_t)b * SEQ * D_MODEL + d;

    float h  = 0.0f;
    float sv = sp[0];
    float xv = xp[0];
    float bv = Bp[0];
    float cv = Cp[0];

    for (int t = 0; t < SEQ; ++t) {
        if ((t & 7) == 0) {                   // global_prefetch_b8 warm-ahead
            __builtin_prefetch(sp + (size_t)(t + 8) * D_MODEL, 0, 1);
            __builtin_prefetch(xp + (size_t)(t + 8) * D_MODEL, 0, 1);
        }
        float svn = 0.0f, xvn = 0.0f, bvn = 0.0f, cvn = 0.0f;
        if (t + 1 < SEQ) {                    // next-step loads before dep chain
            svn = sp[(size_t)(t + 1) * D_MODEL];
            xvn = xp[(size_t)(t + 1) * D_MODEL];
            bvn = Bp[(t + 1) * D_STATE];
            cvn = Cp[(t + 1) * D_STATE];
        }

        h = fmaf(expA * sv, h, bv * sv * xv); // h = A_bar*h + X_bar
        float p = h * cv;
        p += __shfl_xor(p, 8, 16);            // reduce over 16 states
        p += __shfl_xor(p, 4, 16);
        p += __shfl_xor(p, 2, 16);
        p += __shfl_xor(p, 1, 16);
        if (n == 0) yp[(size_t)t * D_MODEL] = p;

        sv = svn; xv = xvn; bv = bvn; cv = cvn;
    }
}

// ---------------------------------------------------------------------------
// g = y * silu(b)   (b = second D_MODEL cols of ab)
// ---------------------------------------------------------------------------
__global__ __launch_bounds__(256)
void gate_kernel(const float* __restrict__ y,
                 const float* __restrict__ ab,
                 float* __restrict__ g)
{
    const int idx = blockIdx.x * blockDim.x + threadIdx.x;
    const int d  = idx % D_MODEL;
    const int bt = idx / D_MODEL;
    const float bval = ab[(size_t)bt * (2 * D_MODEL) + D_MODEL + d];
    g[idx] = y[idx] * silu_f(bval);
}

// ---------------------------------------------------------------------------
// pre = Dp*seq + out; row-wise LayerNorm over D_MODEL; one block per row.
// ---------------------------------------------------------------------------
__global__ __launch_bounds__(256)
void layernorm_kernel(const float* __restrict__ seq,
                      const float* __restrict__ out,
                      const float* __restrict__ Dp,
                      const float* __restrict__ ln_g,
                      const float* __restrict__ ln_b,
                      float* __restrict__ dst)
{
    __shared__ float pre[D_MODEL];
    __shared__ float wsum[8], wsq[8];
    const int row = blockIdx.x;
    const int tid = threadIdx.x;

    float sum = 0.0f, sq = 0.0f;
    for (int d = tid; d < D_MODEL; d += 256) {
        const float v = fmaf(Dp[d], seq[(size_t)row * D_MODEL + d],
                             out[(size_t)row * D_MODEL + d]);
        pre[d] = v;
        sum += v;
        sq  = fmaf(v, v, sq);
    }
#pragma unroll
    for (int o = 16; o > 0; o >>= 1) {
        sum += __shfl_xor(sum, o, 32);
        sq  += __shfl_xor(sq,  o, 32);
    }
    const int wid = tid >> 5, lane = tid & 31;
    if (lane == 0) { wsum[wid] = sum; wsq[wid] = sq; }
    __syncthreads();
    if (tid == 0) {
        float a = 0.0f, c = 0.0f;
#pragma unroll
        for (int i = 0; i < 8; ++i) { a += wsum[i]; c += wsq[i]; }
        wsum[0] = a; wsq[0] = c;
    }
    __syncthreads();
    const float mu  = wsum[0] * (1.0f / D_MODEL);
    const float var = wsq[0] * (1.0f / D_MODEL) - mu * mu;
    const float inv = rsqrtf(var + LN_EPS);
    for (int d = tid; d < D_MODEL; d += 256) {
        dst[(size_t)row * D_MODEL + d] = fmaf(ln_g[d], (pre[d] - mu) * inv, ln_b[d]);
    }
}

// ---------------------------------------------------------------------------
extern "C" void kernel_launch(void* const* d_in, const int* in_sizes, int n_in,
                              void* d_out, int out_size, void* d_ws, size_t ws_size,
                              hipStream_t stream)
{
    const float* seq    = (const float*)d_in[0];
    const float* W_in   = (const float*)d_in[1];
    const float* W_out  = (const float*)d_in[2];
    const float* W_sB   = (const float*)d_in[3];
    const float* W_sC   = (const float*)d_in[4];
    const float* W_sD1  = (const float*)d_in[5];
    const float* W_sD2  = (const float*)d_in[6];
    const float* conv_w = (const float*)d_in[7];
    const float* conv_b = (const float*)d_in[8];
    const float* A      = (const float*)d_in[9];
    const float* Dp     = (const float*)d_in[10];
    const float* ln_g   = (const float*)d_in[11];
    const float* ln_b   = (const float*)d_in[12];
    float* dst = (float*)d_out;

    // Workspace layout (floats)
    float* ws  = (float*)d_ws;
    float* ab  = ws;                                    // [4096, 2048]
    float* x   = ab  + (size_t)M_ROWS * 2 * D_MODEL;    // [4096, 1024]
    float* s   = x   + (size_t)M_ROWS * D_MODEL;        // [4096, 1024]
    float* tmp = s   + (size_t)M_ROWS * D_MODEL;        // [4096, 64]
    float* Bm  = tmp + (size_t)M_ROWS * D_DISCR;        // [4096, 16]
    float* Cm  = Bm  + (size_t)M_ROWS * D_STATE;        // [4096, 16]
    float* y   = Cm  + (size_t)M_ROWS * D_STATE;        // [4096, 1024]
    float* out = y   + (size_t)M_ROWS * D_MODEL;        // [4096, 1024]
    float* g   = x;   // x dead after scan; reuse for gated activation

    const int mt64 = M_ROWS / 64;                       // 64

    // 1) ab = seq @ W_in   [4096x1024 @ 1024x2048]
    wmma_gemm_tiled_kernel<0><<<dim3(2 * D_MODEL / 64, mt64), 256, 0, stream>>>(
        seq, W_in, ab, nullptr, D_MODEL, D_MODEL, 2 * D_MODEL, 2 * D_MODEL);

    // 2) x = silu(causal_conv(a) + bias)
    conv_silu_kernel<<<(M_ROWS * D_MODEL) / 256, 256, 0, stream>>>(ab, conv_w, conv_b, x);

    // 3) tmp = x @ W_sD1   [4096x1024 @ 1024x64]
    wmma_gemm_tiled_kernel<0><<<dim3(D_DISCR / 64, mt64), 256, 0, stream>>>(
        x, W_sD1, tmp, nullptr, D_MODEL, D_MODEL, D_DISCR, D_DISCR);

    // 4) s = softplus(Dp + tmp @ W_sD2)   [4096x64 @ 64x1024]
    wmma_gemm_tiled_kernel<1><<<dim3(D_MODEL / 64, mt64), 256, 0, stream>>>(
        tmp, W_sD2, s, Dp, D_DISCR, D_DISCR, D_MODEL, D_MODEL);

    // 5) Bm = x @ W_sB, Cm = x @ W_sC   [4096x1024 @ 1024x16]
    wmma_gemm_small_kernel<<<dim3(1, M_ROWS / 16), 32, 0, stream>>>(
        x, W_sB, Bm, D_MODEL, D_MODEL, D_STATE, D_STATE);
    wmma_gemm_small_kernel<<<dim3(1, M_ROWS / 16), 32, 0, stream>>>(
        x, W_sC, Cm, D_MODEL, D_MODEL, D_STATE, D_STATE);

    // 6) fused selective scan -> y
    ssm_scan_kernel<<<dim3(D_MODEL / 16, B_SZ), 256, 0, stream>>>(x, s, Bm, Cm, A, y);

    // 7) g = y * silu(b)
    gate_kernel<<<(M_ROWS * D_MODEL) / 256, 256, 0, stream>>>(y, ab, g);

    // 8) out = g @ W_out   [4096x1024 @ 1024x1024]
    wmma_gemm_tiled_kernel<0><<<dim3(D_MODEL / 64, mt64), 256, 0, stream>>>(
        g, W_out, out, nullptr, D_MODEL, D_MODEL, D_MODEL, D_MODEL);

    // 9) residual + LayerNorm -> d_out
    layernorm_kernel<<<M_ROWS, 256, 0, stream>>>(seq, out, Dp, ln_g, ln_b, dst);
}